// MeaMDensity3_34797825032456
// MI455X (gfx1250) — compile-verified
//
#include <hip/hip_runtime.h>
#include <hip/hip_bf16.h>

// ---------------------------------------------------------------------------
// MEAM-style density kernel for MI455X (gfx1250, wave32).
// Counting-sort pairs by destination atom, store per-pair a[16]/r[16] vectors
// densely (padding rows zeroed), then per-atom GEMM via V_WMMA_F32_16X16X4_F32
// with the square + angular-group reduction fused in registers (shfl_xor 16).
// ---------------------------------------------------------------------------

typedef float v2f __attribute__((ext_vector_type(2)));
typedef float v8f __attribute__((ext_vector_type(8)));

#define NWAVE   16
#define NORB    13
#define PI_F    3.14159265358979323846f
#define INV_CUT 0.2f   // 1/5.0

// Compute per-pair a[16] (angular*cij, rows 13..15 = 0) and r[16] (radial).
// Returns false for invalid (shift sentinel) pairs, which contribute nothing.
__device__ __forceinline__ bool pair_compute(
    int p, int npair, int numatom,
    const float* __restrict__ coords, const int* __restrict__ atom_index,
    const float* __restrict__ shifts, const int* __restrict__ species,
    const float* __restrict__ rs, const float* __restrict__ inta,
    const float* __restrict__ params,
    int& idx0_out, float* __restrict__ a, float* __restrict__ r)
{
    int b  = p / npair;
    int lp = p - b * npair;
    float sx = shifts[3 * p + 0];
    float sy = shifts[3 * p + 1];
    float sz = shifts[3 * p + 2];
    if (!(sx > -1.0e9f && sy > -1.0e9f && sz > -1.0e9f)) return false;

    int base = b * 2 * npair;
    int i0 = atom_index[base + lp]         + b * numatom;
    int i1 = atom_index[base + npair + lp] + b * numatom;

    float dx = coords[3 * i0 + 0] - coords[3 * i1 + 0] + sx;
    float dy = coords[3 * i0 + 1] - coords[3 * i1 + 1] + sy;
    float dz = coords[3 * i0 + 2] - coords[3 * i1 + 2] + sz;
    float d  = sqrtf(dx * dx + dy * dy + dz * dz);
    float dinv = (d > 1e-12f) ? (1.0f / d) : 1.0f;

    float fc = 0.5f * (cosf(PI_F * fminf(d * INV_CUT, 1.0f)) + 1.0f);

    int si = species[i1];   // species_i = species[idx1] (radial basis)
    int sj = species[i0];   // species_j = species[idx0]
    float pi_ = (si >= 0) ? params[si] : 0.0f;
    float pj_ = (sj >= 0) ? params[sj] : 0.0f;
    float cij = pi_ * pj_;

    float ux = dx * dinv, uy = dy * dinv, uz = dz * dinv;
    float w1 = fc * ux, w2 = fc * uy, w3 = fc * uz;
    a[0]  = fc * cij;
    a[1]  = w1 * cij;       a[2]  = w2 * cij;       a[3]  = w3 * cij;
    a[4]  = w1 * ux * cij;  a[5]  = w1 * uy * cij;  a[6]  = w1 * uz * cij;
    a[7]  = w2 * ux * cij;  a[8]  = w2 * uy * cij;  a[9]  = w2 * uz * cij;
    a[10] = w3 * ux * cij;  a[11] = w3 * uy * cij;  a[12] = w3 * uz * cij;
    a[13] = 0.0f; a[14] = 0.0f; a[15] = 0.0f;

    int sb = (si >= 0 ? si : 0) * NWAVE;
#pragma unroll
    for (int k = 0; k < NWAVE; ++k) {
        float t = (d - rs[sb + k]) * INV_CUT;
        r[k] = expf(-inta[sb + k] * t * t);
    }
    idx0_out = i0;
    return true;
}

// -------------------- Phase 1: count pairs per destination atom ------------
__global__ void count_kernel(const int* __restrict__ atom_index,
                             const float* __restrict__ shifts,
                             int* __restrict__ counts,
                             int npair, int NP, int numatom)
{
    int p = blockIdx.x * blockDim.x + threadIdx.x;
    if (p >= NP) return;
    float sx = shifts[3 * p + 0];
    float sy = shifts[3 * p + 1];
    float sz = shifts[3 * p + 2];
    if (!(sx > -1.0e9f && sy > -1.0e9f && sz > -1.0e9f)) return;
    int b  = p / npair;
    int lp = p - b * npair;
    int i0 = atom_index[b * 2 * npair + lp] + b * numatom;
    atomicAdd(&counts[i0], 1);
}

// -------------------- Phase 2: exclusive scan of padded counts -------------
// Single block of 1024 threads; each thread owns a contiguous chunk.
__global__ void scan_kernel(const int* __restrict__ counts,
                            int* __restrict__ offsets,
                            int* __restrict__ cursors,
                            int natoms)
{
    __shared__ int lds[1024];
    int t = threadIdx.x;
    int items = (natoms + 1023) >> 10;   // <= 8 for natoms <= 8192
    int local[8];
    int sum = 0;
#pragma unroll
    for (int i = 0; i < 8; ++i) {
        if (i >= items) break;
        int idx = t * items + i;
        int pc = 0;
        if (idx < natoms) pc = (counts[idx] + 3) & ~3;   // pad K to multiple of 4
        local[i] = sum;
        sum += pc;
    }
    lds[t] = sum;
    __syncthreads();
    for (int offd = 1; offd < 1024; offd <<= 1) {
        int v = lds[t];
        int add = (t >= offd) ? lds[t - offd] : 0;
        __syncthreads();
        lds[t] = v + add;
        __syncthreads();
    }
    int base = (t == 0) ? 0 : lds[t - 1];
#pragma unroll
    for (int i = 0; i < 8; ++i) {
        if (i >= items) break;
        int idx = t * items + i;
        if (idx < natoms) {
            int o = base + local[i];
            offsets[idx] = o;
            cursors[idx] = o;
        }
    }
}

// -------------------- Phase 2b: zero the K-padding rows of A and R ---------
// One thread per atom; at most 3 padding slots each -> guard-free GEMM loads.
__global__ void pad_zero_kernel(const int* __restrict__ counts,
                                const int* __restrict__ offsets,
                                float* __restrict__ A, float* __restrict__ R,
                                int natoms)
{
    int a = blockIdx.x * blockDim.x + threadIdx.x;
    if (a >= natoms) return;
    int cnt = counts[a];
    int off = offsets[a];
    int pc  = (cnt + 3) & ~3;
    float4 z = make_float4(0.f, 0.f, 0.f, 0.f);
    for (int s = cnt; s < pc; ++s) {
        float4* Ap = (float4*)(A + (size_t)(off + s) * 16);
        float4* Rp = (float4*)(R + (size_t)(off + s) * 16);
#pragma unroll
        for (int q = 0; q < 4; ++q) { Ap[q] = z; Rp[q] = z; }
    }
}

// -------------------- Phase 3: scatter pair vectors into sorted slots ------
__global__ void scatter_kernel(const float* __restrict__ coords,
                               const int* __restrict__ atom_index,
                               const float* __restrict__ shifts,
                               const int* __restrict__ species,
                               const float* __restrict__ rs,
                               const float* __restrict__ inta,
                               const float* __restrict__ params,
                               int* __restrict__ cursors,
                               float* __restrict__ A, float* __restrict__ R,
                               int npair, int NP, int numatom)
{
    int p = blockIdx.x * blockDim.x + threadIdx.x;
    if (p >= NP) return;
    __attribute__((aligned(16))) float a[16];
    __attribute__((aligned(16))) float r[16];
    int i0;
    if (!pair_compute(p, npair, numatom, coords, atom_index, shifts, species,
                      rs, inta, params, i0, a, r)) return;
    int slot = atomicAdd(&cursors[i0], 1);
    float4* Ap = (float4*)(A + (size_t)slot * 16);
    float4* Rp = (float4*)(R + (size_t)slot * 16);
    const float4* av = (const float4*)a;
    const float4* rv = (const float4*)r;
#pragma unroll
    for (int q = 0; q < 4; ++q) { Ap[q] = av[q]; Rp[q] = rv[q]; }
}

// -------------------- Phase 4: per-atom WMMA GEMM + fused square/reduce ----
// One wave32 per atom. A tile = 16x4 f32 (4 pairs' a-vectors), B = 4x16
// (4 pairs' r-vectors), C accumulates part[atom] (16x16, rows 13..15 zero).
// A layout: v0 = A[M=lane%16][K = lane<16 ? 0 : 2], v1 = K=1/3.
// B layout: v0 = B[K = lane<16 ? 0 : 2][N=lane%16], v1 = K=1/3.
// C layout: VGPR v holds M=v (lanes 0-15) and M=v+8 (lanes 16-31), N=lane%16.
// All K-padding slots are zero, so every load is unconditional.
__global__ void __launch_bounds__(256)
gemm_density_kernel(const float* __restrict__ A, const float* __restrict__ R,
                    const int* __restrict__ counts,
                    const int* __restrict__ offsets,
                    float* __restrict__ out, int natoms)
{
    int wave = threadIdx.x >> 5;
    int atom = blockIdx.x * 8 + wave;
    if (atom >= natoms) return;
    int lane = threadIdx.x & 31;
    int half = lane >> 4;
    int l    = lane & 15;

    int pc  = (counts[atom] + 3) & ~3;   // padded K extent (padding zeroed)
    int off = offsets[atom];

    // Per-lane base pointers: element for K-group c, sub-K j lives at [(c+j)*16].
    const float* __restrict__ Ap = A + ((size_t)off + 2 * half) * 16 + l;
    const float* __restrict__ Rp = R + ((size_t)off + 2 * half) * 16 + l;

    v8f C = {};
    int c = 0;
    // Main loop: 16 pairs (4 WMMAs) per iteration; 16 loads per lane issued up front.
    for (; c + 16 <= pc; c += 16) {
        float aE[8], bE[8];
#pragma unroll
        for (int g = 0; g < 4; ++g) {
            aE[2 * g]     = Ap[(size_t)(c + 4 * g) * 16];
            aE[2 * g + 1] = Ap[(size_t)(c + 4 * g + 1) * 16];
            bE[2 * g]     = Rp[(size_t)(c + 4 * g) * 16];
            bE[2 * g + 1] = Rp[(size_t)(c + 4 * g + 1) * 16];
        }
#pragma unroll
        for (int g = 0; g < 4; ++g) {
            v2f av = { aE[2 * g], aE[2 * g + 1] };
            v2f bv = { bE[2 * g], bE[2 * g + 1] };
            C = __builtin_amdgcn_wmma_f32_16x16x4_f32(
                    false, av, false, bv, (short)0, C, false, false);
        }
    }
    // Tail: remaining groups of 4 (zero-padded, unguarded).
    for (; c < pc; c += 4) {
        v2f av = { Ap[(size_t)c * 16], Ap[(size_t)(c + 1) * 16] };
        v2f bv = { Rp[(size_t)c * 16], Rp[(size_t)(c + 1) * 16] };
        C = __builtin_amdgcn_wmma_f32_16x16x4_f32(
                false, av, false, bv, (short)0, C, false, false);
    }

    // Square + reduce over angular groups {m=0}, {m=1..3}, {m=4..12}.
    float sq0 = C[0] * C[0], sq1 = C[1] * C[1], sq2 = C[2] * C[2],
          sq3 = C[3] * C[3], sq4 = C[4] * C[4], sq5 = C[5] * C[5],
          sq6 = C[6] * C[6], sq7 = C[7] * C[7];
    float g0 = sq0;                 // valid on half 0 (m=0)
    float g1 = sq1 + sq2 + sq3;     // valid on half 0 (m=1..3)
    // half0 contributes m=4..7; half1 contributes m=8..12 (13..15 are zero).
    float u  = half ? (sq0 + sq1 + sq2 + sq3 + sq4) : (sq4 + sq5 + sq6 + sq7);
    float g2 = u + __shfl_xor(u, 16, 32);

    if (half == 0) {
        size_t ob = (size_t)atom * 48;
        out[ob + l]      = g0;
        out[ob + 16 + l] = g1;
        out[ob + 32 + l] = g2;
    }
}

// -------------------- Fallback path (small workspace): direct atomics ------
__global__ void pair_atomic_kernel(const float* __restrict__ coords,
                                   const int* __restrict__ atom_index,
                                   const float* __restrict__ shifts,
                                   const int* __restrict__ species,
                                   const float* __restrict__ rs,
                                   const float* __restrict__ inta,
                                   const float* __restrict__ params,
                                   float* __restrict__ part,
                                   int npair, int NP, int numatom)
{
    int p = blockIdx.x * blockDim.x + threadIdx.x;
    if (p >= NP) return;
    float a[16], r[16];
    int i0;
    if (!pair_compute(p, npair, numatom, coords, atom_index, shifts, species,
                      rs, inta, params, i0, a, r)) return;
    float* base = part + (size_t)i0 * 256;
#pragma unroll
    for (int j = 0; j < NORB; ++j)
#pragma unroll
        for (int k = 0; k < NWAVE; ++k)
            atomicAdd(&base[j * 16 + k], a[j] * r[k]);
}

__global__ void finalize_kernel(const float* __restrict__ part,
                                float* __restrict__ out, int natoms)
{
    int tid = blockIdx.x * blockDim.x + threadIdx.x;
    if (tid >= natoms * NWAVE) return;
    int atom = tid >> 4;
    int k    = tid & 15;
    const float* base = part + (size_t)atom * 256 + k;
    float g0 = base[0] * base[0];
    float g1 = 0.0f, g2 = 0.0f;
#pragma unroll
    for (int j = 1; j < 4; ++j)  { float v = base[j * 16]; g1 += v * v; }
#pragma unroll
    for (int j = 4; j < 13; ++j) { float v = base[j * 16]; g2 += v * v; }
    size_t ob = (size_t)atom * 48;
    out[ob + k]      = g0;
    out[ob + 16 + k] = g1;
    out[ob + 32 + k] = g2;
}

// ---------------------------------------------------------------------------
extern "C" void kernel_launch(void* const* d_in, const int* in_sizes, int n_in,
                              void* d_out, int out_size, void* d_ws, size_t ws_size,
                              hipStream_t stream)
{
    const float* coords     = (const float*)d_in[0];   // (nbatch, numatom, 3)
    const int*   atom_index = (const int*)  d_in[2];   // (nbatch, 2, npair)
    const float* shifts     = (const float*)d_in[3];   // (nbatch, npair, 3)
    const int*   species    = (const int*)  d_in[4];   // (totnatom,)
    const float* rs         = (const float*)d_in[5];   // (natomtype, nwave)
    const float* inta       = (const float*)d_in[6];   // (natomtype, nwave)
    const float* params     = (const float*)d_in[7];   // (natomtype,)
    float* out = (float*)d_out;

    int nbatch   = in_sizes[1];                 // numatoms is (nbatch, 1)
    int totnatom = in_sizes[0] / 3;
    int numatom  = totnatom / nbatch;
    int npair    = in_sizes[3] / (3 * nbatch);  // pairs per batch
    int NP       = nbatch * npair;              // total pairs

    const int TB = 256;
    int pairBlocks = (NP + TB - 1) / TB;

    // Sorted + WMMA path workspace: counts/offsets/cursors + A + R slot arrays.
    size_t maxslots = (size_t)NP + 3ull * totnatom + 16;
    size_t intBytes = (size_t)totnatom * 3 * sizeof(int);
    size_t matBytes = maxslots * 16 * sizeof(float);
    size_t needSorted = intBytes + 2 * matBytes;

    if (ws_size >= needSorted) {
        int*   counts  = (int*)d_ws;
        int*   offsets = counts + totnatom;
        int*   cursors = offsets + totnatom;
        float* A       = (float*)(cursors + totnatom);
        float* R       = A + maxslots * 16;

        hipMemsetAsync(counts, 0, (size_t)totnatom * sizeof(int), stream);
        count_kernel<<<pairBlocks, TB, 0, stream>>>(atom_index, shifts, counts,
                                                    npair, NP, numatom);
        scan_kernel<<<1, 1024, 0, stream>>>(counts, offsets, cursors, totnatom);
        pad_zero_kernel<<<(totnatom + TB - 1) / TB, TB, 0, stream>>>(
            counts, offsets, A, R, totnatom);
        scatter_kernel<<<pairBlocks, TB, 0, stream>>>(coords, atom_index, shifts,
                                                      species, rs, inta, params,
                                                      cursors, A, R,
                                                      npair, NP, numatom);
        gemm_density_kernel<<<(totnatom + 7) / 8, TB, 0, stream>>>(
            A, R, counts, offsets, out, totnatom);
    } else {
        // Fallback: direct float atomics into a 16x16-per-atom part buffer.
        float* part = (float*)d_ws;
        hipMemsetAsync(part, 0, (size_t)totnatom * 256 * sizeof(float), stream);
        pair_atomic_kernel<<<pairBlocks, TB, 0, stream>>>(coords, atom_index,
                                                          shifts, species, rs,
                                                          inta, params, part,
                                                          npair, NP, numatom);
        int fthreads = totnatom * NWAVE;
        finalize_kernel<<<(fthreads + TB - 1) / TB, TB, 0, stream>>>(part, out,
                                                                     totnatom);
    }
}